// posnegECELoss_47923245089174
// MI455X (gfx1250) — compile-verified
//
#include <hip/hip_runtime.h>

// posnegECELoss for MI455X (gfx1250, wave32).
// logits: [50000, 1000] f32, labels: [50000] i32.
// out: pos_sce[1000] ++ neg_sce[1000] ++ classes_acc[1000]  (3000 f32)
// ws (floats): gCnt[15000] gConf[15000] gCorr[15000] aCorr[1000] aCnt[1000]
//   = 47000 floats = 188000 bytes required.

#define N_ROWS  50000
#define N_CLS   1000
#define NBINS   15
#define CB      (N_CLS * NBINS)

#define WAVES   16
#define BLOCK   (WAVES * 32)
#define NBLOCKS 512
#define JITER   32   // ceil(N_CLS / 32)

#if defined(__has_builtin)
# if __has_builtin(__builtin_amdgcn_global_load_async_to_lds_b128) && \
     __has_builtin(__builtin_amdgcn_s_wait_asynccnt)
#  define USE_ASYNC 1
# endif
#endif
#ifndef USE_ASYNC
# define USE_ASYNC 0
#endif

#if USE_ASYNC
typedef int v4i __attribute__((ext_vector_type(4)));
typedef __attribute__((address_space(1))) v4i* gptr_v4i;   // global int4*
typedef __attribute__((address_space(3))) v4i* lptr_v4i;   // LDS int4*

// Stream one 1000-float row (250 x b128) into an LDS staging row.
// 8 async wave-instructions per row (last one partially masked).
__device__ __forceinline__ void issue_row_async(const float* __restrict__ grow,
                                                float* dstRow, int lane) {
#pragma unroll
  for (int k = 0; k < 8; ++k) {
    const int i4 = lane + 32 * k;       // float4 index, 250 total
    if (i4 < 250) {
      float* gsrc = const_cast<float*>(grow + 4 * i4);
      __builtin_amdgcn_global_load_async_to_lds_b128(
          (gptr_v4i)gsrc,
          (lptr_v4i)(dstRow + 4 * i4),
          0, 0);
    }
  }
}
#endif

__global__ __launch_bounds__(BLOCK) void sce_zero_ws(float* __restrict__ ws, int n) {
  const int i = blockIdx.x * blockDim.x + threadIdx.x;
  if (i < n) ws[i] = 0.0f;
}

__global__ __launch_bounds__(BLOCK) void sce_accum(const float* __restrict__ logits,
                                                   const int* __restrict__ labels,
                                                   float* __restrict__ ws) {
  float* gCnt  = ws;
  float* gConf = ws + CB;
  float* gCorr = ws + 2 * CB;
  float* aCorr = ws + 3 * CB;
  float* aCnt  = ws + 3 * CB + N_CLS;

  // Per-block (class,bin) histogram privatized in LDS: 3 * 60000 B = 180 KB.
  __shared__ float    s_conf[CB];
  __shared__ float    s_corr[CB];
  __shared__ unsigned s_cnt [CB];
#if USE_ASYNC
  // Double-buffered per-wave row staging: 16 waves * 2 * 4 KB = 128 KB.
  __shared__ float    s_stage[WAVES][2][1024];
#endif

  const int tid  = threadIdx.x;
  const int lane = tid & 31;
  const int wid  = tid >> 5;

  for (int i = tid; i < CB; i += BLOCK) { s_conf[i] = 0.0f; s_corr[i] = 0.0f; s_cnt[i] = 0u; }
  __syncthreads();

  const int totalWaves = (int)gridDim.x * WAVES;
  const int gw = (int)blockIdx.x * WAVES + wid;   // one wave == one row

#if USE_ASYNC
  int parity = 0;
  if (gw < N_ROWS)
    issue_row_async(logits + (size_t)gw * N_CLS, &s_stage[wid][0][0], lane);
#endif

  for (int n = gw; n < N_ROWS; n += totalWaves) {
#if USE_ASYNC
    const int nn = n + totalWaves;
    if (nn < N_ROWS) {
      issue_row_async(logits + (size_t)nn * N_CLS, &s_stage[wid][parity ^ 1][0], lane);
      __builtin_amdgcn_s_wait_asynccnt(8);   // 8 outstanding = next row; current row done
    } else {
      __builtin_amdgcn_s_wait_asynccnt(0);
    }
    const float* row = &s_stage[wid][parity][0];
    parity ^= 1;
#else
    const float* row = logits + (size_t)n * N_CLS;
    if (n + totalWaves < N_ROWS)   // per-lane addresses cover the whole next 4 KB row
      __builtin_prefetch(logits + (size_t)(n + totalWaves) * N_CLS + lane * 32, 0, 1);
#endif

    // Lane L holds classes c = L + 32*j, j = 0..31 (c < 1000).
    float x[JITER];
#pragma unroll
    for (int j = 0; j < JITER; ++j) {
      const int c = lane + 32 * j;
      x[j] = (c < N_CLS) ? row[c] : -3.0e38f;
    }

    // Row max + argmax (ties -> lowest class, matching jnp.argmax).
    float bestv = x[0];
    int   bestc = lane;
#pragma unroll
    for (int j = 1; j < JITER; ++j) {
      const int c = lane + 32 * j;
      if (x[j] > bestv) { bestv = x[j]; bestc = c; }
    }
#pragma unroll
    for (int off = 16; off >= 1; off >>= 1) {
      const float ov = __shfl_xor(bestv, off, 32);
      const int   oc = __shfl_xor(bestc, off, 32);
      if (ov > bestv || (ov == bestv && oc < bestc)) { bestv = ov; bestc = oc; }
    }
    const float m = bestv;

    // Sum of exp (wave32 butterfly).
    float ssum = 0.0f;
#pragma unroll
    for (int j = 0; j < JITER; ++j) ssum += __expf(x[j] - m);
#pragma unroll
    for (int off = 16; off >= 1; off >>= 1) ssum += __shfl_xor(ssum, off, 32);
    const float inv = 1.0f / ssum;

    const int lab = labels[n];

    // Scatter into the privatized (class,bin) histogram via LDS atomics.
    // Distinct class per lane -> distinct LDS address per lane.
#pragma unroll
    for (int j = 0; j < JITER; ++j) {
      const int c = lane + 32 * j;
      if (c < N_CLS) {
        const float p = __expf(x[j] - m) * inv;
        if (p > 0.0f) {                 // bin b covers (b/15, (b+1)/15]
          const float t = p * (float)NBINS;
          int b = (int)t;
          if (b > 0 && (float)b == t) b -= 1;
          if (b > NBINS - 1) b = NBINS - 1;
          const int idx = c * NBINS + b;
          atomicAdd(&s_conf[idx], p);   // ds_add_f32
          atomicAdd(&s_cnt[idx], 1u);   // ds_add_u32
          if (c == lab) atomicAdd(&s_corr[idx], 1.0f);
        }
      }
    }

    if (lane == 0) {                    // per-class top-1 accuracy tallies
      atomicAdd(&aCnt[lab], 1.0f);
      if (bestc == lab) atomicAdd(&aCorr[lab], 1.0f);
    }
  }

  // Flush only nonzero cells to global (float atomics; counts < 2^24 stay exact).
  __syncthreads();
  for (int i = tid; i < CB; i += BLOCK) {
    const unsigned cv = s_cnt[i];
    if (cv) {
      atomicAdd(&gCnt[i], (float)cv);
      atomicAdd(&gConf[i], s_conf[i]);
      const float cr = s_corr[i];
      if (cr != 0.0f) atomicAdd(&gCorr[i], cr);
    }
  }
}

__global__ void sce_finalize(const float* __restrict__ ws, float* __restrict__ out) {
  const int c = blockIdx.x * blockDim.x + threadIdx.x;
  if (c >= N_CLS) return;
  const float* gCnt  = ws;
  const float* gConf = ws + CB;
  const float* gCorr = ws + 2 * CB;
  const float* aCorr = ws + 3 * CB;
  const float* aCnt  = ws + 3 * CB + N_CLS;

  float pos = 0.0f, neg = 0.0f;
#pragma unroll
  for (int b = 0; b < NBINS; ++b) {
    const int idx = c * NBINS + b;
    const float cnt = gCnt[idx];
    if (cnt > 0.0f) {
      const float diff = (gConf[idx] - gCorr[idx]) / cnt;      // avg_conf - acc
      const float contrib = fabsf(diff) * (cnt / (float)N_ROWS);
      if (diff > 0.0f) pos += contrib; else neg += contrib;
    }
  }
  out[c]             = pos;
  out[N_CLS + c]     = neg;
  out[2 * N_CLS + c] = aCorr[c] / fmaxf(aCnt[c], 1.0f);
}

extern "C" void kernel_launch(void* const* d_in, const int* in_sizes, int n_in,
                              void* d_out, int out_size, void* d_ws, size_t ws_size,
                              hipStream_t stream) {
  (void)in_sizes; (void)n_in; (void)out_size; (void)ws_size;
  const float* logits = (const float*)d_in[0];
  const int*   labels = (const int*)d_in[1];
  float* ws  = (float*)d_ws;
  float* out = (float*)d_out;

  const int totalWs = 3 * CB + 2 * N_CLS;   // 47000 floats
  sce_zero_ws<<<(totalWs + BLOCK - 1) / BLOCK, BLOCK, 0, stream>>>(ws, totalWs);
  sce_accum<<<NBLOCKS, BLOCK, 0, stream>>>(logits, labels, ws);
  sce_finalize<<<(N_CLS + 255) / 256, 256, 0, stream>>>(ws, out);
}